// GAE_18365280158228
// MI455X (gfx1250) — compile-verified
//
#include <hip/hip_runtime.h>

// ---------------- problem constants ----------------
#define GAE_T   1024
#define GAE_B   16384
#define GAMMA_F 0.99f
#define CF      ((float)(0.99 * 0.97))   // jnp.float32(GAMMA*LAMBDA): double product -> f32
#define CT      16                       // timesteps per WMMA chunk
#define NC      (GAE_T / CT)             // 64 chunks
#define NBT     (GAE_B / 16)             // 1024 column tiles
#define NTILES  (NC * NBT)               // 65536 wave-tiles

typedef __attribute__((ext_vector_type(2))) float v2f;
typedef __attribute__((ext_vector_type(8))) float v8f;

// ---- compile-time tables, laid out in WMMA A-operand register order ----
// A 16x4 layout (ISA 7.12.2): lane m = lane&15; vgpr0 holds K0 = 4s+2*half,
// vgpr1 holds K0+1.  L[m][k] = (k>=m) ? coef^(k-m) : 0  (upper triangular).
struct alignas(8) F2 { float x, y; };
struct GaeTab {
  F2    a[4][32];     // A operand for WMMA step s, per lane: one b64 load each
  float cpow[17];     // coef^e, e in [0,16] (sequential-multiply rounding)
};
constexpr GaeTab make_tab() {
  GaeTab t{};
  float p = 1.0f;
  for (int i = 0; i <= 16; ++i) { t.cpow[i] = p; p = p * CF; }
  for (int s = 0; s < 4; ++s)
    for (int lane = 0; lane < 32; ++lane) {
      const int n    = lane & 15;
      const int half = lane >> 4;
      const int k0   = 4 * s + 2 * half;
      t.a[s][lane].x = (k0 - n     >= 0) ? t.cpow[k0 - n]     : 0.0f;
      t.a[s][lane].y = (k0 + 1 - n >= 0) ? t.cpow[k0 + 1 - n] : 0.0f;
    }
  return t;
}
__constant__ GaeTab GTAB = make_tab();
#define CF16C (GTAB.cpow[16])

// One wave == one 16(t) x 16(b) tile.  D = L * delta + C via 4 chained
// V_WMMA_F32_16X16X4_F32 (full fp32 precision).
//
// MODE 0: C = 0, write only the chunk-local sum at m=0 into out[(NC,B)].
// MODE 1: C[m][n] = coef^(16-m) * carry[c][b] (true adv at chunk end),
//         write the final adv tile into out[(T,B)].
template <int MODE>
__global__ __launch_bounds__(256) void gae_tile_kernel(
    const float* __restrict__ value,   // (T+1, B)
    const float* __restrict__ reward,  // (T,   B)
    float* __restrict__ out,
    const float* __restrict__ carry)   // (NC, B) -- MODE 1 only
{
  const int lane = threadIdx.x & 31;
  const int wid  = threadIdx.x >> 5;
  const int tile = blockIdx.x * 8 + wid;     // < NTILES
  const int c    = tile >> 10;               // chunk index   [0, 64)
  const int jb   = tile & 1023;              // column tile   [0, 1024)
  const int n    = lane & 15;                // tile column
  const int half = lane >> 4;                // 0: lanes 0-15, 1: lanes 16-31
  const int b    = (jb << 4) + n;            // global column
  const int t0   = c << 4;                   // chunk start timestep

  // ----- per-lane delta rows for the B operand -----
  // B 4x16 layout (mirrors documented A layout): vgpr0/vgpr1 hold rows
  // K = 2*half + {0,1} (+4 per WMMA step), column N = lane&15.
  const float* vcol = value  + b;
  const float* rcol = reward + b;
  float dly[8];
#pragma unroll
  for (int s = 0; s < 4; ++s) {
    const int k  = 4 * s + 2 * half;         // first row of this pair
    const int t  = t0 + k;
    const float v0 = vcol[t * GAE_B];
    const float v1 = vcol[(t + 1) * GAE_B];  // shared between the two deltas
    const float v2 = vcol[(t + 2) * GAE_B];
    const float r0 = rcol[t * GAE_B];
    const float r1 = rcol[(t + 1) * GAE_B];
    dly[2 * s]     = fmaf(GAMMA_F, v1, r0) - v0;   // delta[t]
    dly[2 * s + 1] = fmaf(GAMMA_F, v2, r1) - v1;   // delta[t+1]
  }

  // ----- accumulator init (C matrix; vgpr j holds row m = j + 8*half) -----
  v8f acc;
  if (MODE == 0) {
#pragma unroll
    for (int j = 0; j < 8; ++j) acc[j] = 0.0f;
  } else {
    const float cin = carry[c * GAE_B + b];         // adv[(c+1)*16][b]
#pragma unroll
    for (int j = 0; j < 8; ++j) {
      const int m = j + 8 * half;                   // index in [1,16]: no guard
      acc[j] = GTAB.cpow[16 - m] * cin;             // coef^(16-m) * carry
    }
  }

  // ----- 4 chained V_WMMA_F32_16X16X4_F32: D = L * delta + C -----
#pragma unroll
  for (int s = 0; s < 4; ++s) {
    const F2 af = GTAB.a[s][lane];                  // one aligned b64 load
    v2f a;  a.x  = af.x;      a.y  = af.y;
    v2f bb; bb.x = dly[2*s];  bb.y = dly[2*s + 1];
    acc = __builtin_amdgcn_wmma_f32_16x16x4_f32(false, a, false, bb,
                                                (short)0, acc, false, false);
  }

  // ----- epilogue -----
  if (MODE == 0) {
    if (lane < 16)                                  // vgpr0, lanes 0-15 = row m=0
      out[c * GAE_B + b] = acc[0];
  } else {
#pragma unroll
    for (int j = 0; j < 8; ++j) {
      const int t = t0 + j + 8 * half;
      out[t * GAE_B + b] = acc[j];
    }
  }
}

// Sequential scan over the 64 chunk summaries (L2-resident, 4 MB each way):
//   carry[c] = A_{c+1};  A_c = s[c] + coef^16 * A_{c+1};  A_{NC} = 0
__global__ __launch_bounds__(256) void gae_carry_scan(
    const float* __restrict__ sums, float* __restrict__ carry)
{
  const int b = blockIdx.x * 256 + threadIdx.x;
  const float cf16 = CF16C;
  float acc = 0.0f;
#pragma unroll 8
  for (int c = NC - 1; c >= 0; --c) {
    carry[c * GAE_B + b] = acc;
    acc = fmaf(cf16, acc, sums[c * GAE_B + b]);
  }
}

extern "C" void kernel_launch(void* const* d_in, const int* in_sizes, int n_in,
                              void* d_out, int out_size, void* d_ws, size_t ws_size,
                              hipStream_t stream) {
  (void)in_sizes; (void)n_in; (void)out_size; (void)ws_size;
  const float* value  = (const float*)d_in[0];   // (T+1, B) f32
  const float* reward = (const float*)d_in[1];   // (T,   B) f32
  float* adv   = (float*)d_out;                  // (T, B) f32
  float* sums  = (float*)d_ws;                   // NC*B floats = 4 MB
  float* carry = sums + (size_t)NC * GAE_B;      // NC*B floats = 4 MB

  const dim3 blk(256);
  // Pass 1: chunk-local sums (reads 128 MB from HBM, warms 192 MB L2)
  gae_tile_kernel<0><<<dim3(NTILES / 8), blk, 0, stream>>>(value, reward, sums, nullptr);
  // Pass 2: 64-step carry scan per column (L2-resident)
  gae_carry_scan<<<dim3(GAE_B / 256), blk, 0, stream>>>(sums, carry);
  // Pass 3: recompute tiles with carry folded into the WMMA accumulator,
  //         write final adv (inputs re-read from L2; 64 MB HBM write)
  gae_tile_kernel<1><<<dim3(NTILES / 8), blk, 0, stream>>>(value, reward, adv, carry);
}